// VectorQuantizerEMA_72722386256094
// MI455X (gfx1250) — compile-verified
//
#include <hip/hip_runtime.h>

typedef float v2f __attribute__((ext_vector_type(2)));
typedef float v8f __attribute__((ext_vector_type(8)));
typedef unsigned int u32x4 __attribute__((ext_vector_type(4)));
typedef int i32x8 __attribute__((ext_vector_type(8)));
typedef int i32x4 __attribute__((ext_vector_type(4)));

#define VOCAB   8192
#define EMBED   256
#define HW      1024          // 32*32
#define MROWS   8192          // B*H*W
#define MT      128           // rows per block
#define VSLICES 8             // grid.y: V split
#define VCHUNK  64            // codebook rows per TDM chunk
#define NCHUNK  16            // (VOCAB/VSLICES)/VCHUNK
#define LSTRIDE 260           // 256 + 4 pad; stride % 64 == 4 -> conflict-free frag reads
#define BCHUNKF (VCHUNK * LSTRIDE)   // floats per B buffer

// workspace layout (bytes)
#define WS_NORM_OFF   0         // 8192 floats
#define WS_MIN_OFF    32768     // 8192 u64
#define WS_LOSSP_OFF  98304     // 512 floats

// d_out layout (floats): z_q4 | indices | loss | perplexity
#define OUT_IDX_OFF   2097152
#define OUT_LOSS_OFF  2105344
#define OUT_PERP_OFF  2105345

// ---------------- Kernel 0: codebook row norms + argmin-slot init ----------------
__global__ __launch_bounds__(256)
void vq_init_normsq(const float* __restrict__ cb,
                    float* __restrict__ norm_e,
                    unsigned long long* __restrict__ minslot) {
  const int t = threadIdx.x;
  const int lane = t & 31;
  const int wave = t >> 5;
  const int v = blockIdx.x * 8 + wave;
  const float* row = cb + (size_t)v * EMBED;
  float4 a = *(const float4*)(row + lane * 8);
  float4 b = *(const float4*)(row + lane * 8 + 4);
  float s = a.x*a.x + a.y*a.y + a.z*a.z + a.w*a.w
          + b.x*b.x + b.y*b.y + b.z*b.z + b.w*b.w;
  #pragma unroll
  for (int m = 16; m >= 1; m >>= 1) s += __shfl_xor(s, m, 32);
  if (lane == 0) {
    norm_e[v] = s;
    minslot[v] = ~0ull;
  }
}

// Issue one TDM load: VCHUNK codebook rows (256 f32 each) -> LDS, with 4-DWORD
// padding inserted after every 256 DWORDs so rows land at stride LSTRIDE=260.
__device__ __forceinline__ void tdm_load_chunk(unsigned ldsOff, unsigned long long ga) {
  u32x4 g0;
  g0[0] = 1u;                                   // count=1, is_restore=0, gather off
  g0[1] = ldsOff;                               // lds_addr (bytes)
  g0[2] = (unsigned)ga;                         // global_addr[31:0]
  g0[3] = (unsigned)((ga >> 32) & 0x1FFFFFFull) // global_addr[56:32]
        | 0x80000000u;                          // type=2 ("image") in bits 127:126
  i32x8 g1;
  g1[0] = (2 << 16)      // data_size = 4B
        | (1 << 20)      // pad_enable
        | (7 << 22)      // pad_interval: 256 DWORDs
        | (3 << 25);     // pad_amount: 4 DWORDs
  g1[1] = 256 << 16;     // tensor_dim0[15:0] = 256 at bits 63:48
  g1[2] = VCHUNK << 16;  // tensor_dim1[15:0] = 64 at bits 95:80
  g1[3] = 256 << 16;     // tile_dim0 = 256 at bits 127:112
  g1[4] = VCHUNK;        // tile_dim1 = 64 (tile_dim2 = 0)
  g1[5] = 256;           // tensor_dim0_stride = 256 elements
  g1[6] = 0;
  g1[7] = 0;
  i32x4 gz  = {0, 0, 0, 0};                     // 2-D tensor: groups 2/3 unused
  i32x8 gz8 = {0, 0, 0, 0, 0, 0, 0, 0};         // clang-23 6-arg form: extra desc operand
  __builtin_amdgcn_tensor_load_to_lds(g0, g1, gz, gz, gz8, 0);
}

// ---------------- Kernel 1: WMMA scores, TDM-double-buffered B, fused argmin ----------------
__global__ __launch_bounds__(256)
void vq_scores(const float* __restrict__ z,
               const float* __restrict__ cb,
               const float* __restrict__ norm_e,
               unsigned long long* __restrict__ minslot) {
  __shared__ float As[MT * LSTRIDE];        // 133,120 B
  __shared__ float Bs[2 * BCHUNKF];         // 2 x 66,560 B

  const int t     = threadIdx.x;
  const int lane  = t & 31;
  const int wave  = t >> 5;
  const int mblk  = blockIdx.x * MT;
  const int vbase = blockIdx.y * (VOCAB / VSLICES);
  const int bb    = mblk >> 10;             // batch (MT divides HW)
  const int n0    = mblk & (HW - 1);

  const unsigned long long cbAddr = (unsigned long long)(const void*)cb;
  const unsigned bsOff0 = (unsigned)(unsigned long long)(const void*)&Bs[0];
  const unsigned bsOff1 = (unsigned)(unsigned long long)(const void*)&Bs[BCHUNKF];

  // Kick off TDM for chunk 0 while all waves stage the A tile.
  if (wave == 0) {
    tdm_load_chunk(bsOff0, cbAddr + (unsigned long long)vbase * (EMBED * 4));
  }

  // Stage A tile: z_flat[m][c] = z[b][c][n]; rows contiguous in n.
  const size_t zb = (size_t)bb * EMBED * HW + n0;
  #pragma unroll
  for (int it = 0; it < 32; ++it) {
    int e  = it * 256 + t;
    int c  = e >> 5;                        // 0..255
    int r4 = e & 31;                        // rows r4*4 .. +3
    float4 vz = *(const float4*)(z + zb + (size_t)c * HW + r4 * 4);
    As[(r4*4+0)*LSTRIDE + c] = vz.x;
    As[(r4*4+1)*LSTRIDE + c] = vz.y;
    As[(r4*4+2)*LSTRIDE + c] = vz.z;
    As[(r4*4+3)*LSTRIDE + c] = vz.w;
  }
  if (wave == 0) __builtin_amdgcn_s_wait_tensorcnt(0);
  __syncthreads();

  const int wr   = wave >> 2;               // 0..1 -> row-tiles wr*4 .. +3
  const int wc   = wave & 3;                // 0..3 -> col-tile
  const int half = lane >> 4;
  const int l15  = lane & 15;

  unsigned long long best[4][8];
  #pragma unroll
  for (int rt = 0; rt < 4; ++rt)
    #pragma unroll
    for (int i = 0; i < 8; ++i) best[rt][i] = ~0ull;

  #pragma unroll 1
  for (int ch = 0; ch < NCHUNK; ++ch) {
    const int vchunk = vbase + ch * VCHUNK;
    // Prefetch next chunk into the other buffer (single TDM issue, wave 0).
    if (wave == 0 && (ch + 1) < NCHUNK) {
      tdm_load_chunk((ch & 1) ? bsOff0 : bsOff1,
                     cbAddr + (unsigned long long)(vchunk + VCHUNK) * (EMBED * 4));
    }
    const float* Bsc = Bs + (ch & 1) * BCHUNKF;
    const unsigned vcol = (unsigned)(vchunk + wc * 16 + l15);
    const float ne = norm_e[vcol];

    v8f acc[4] = {};
    #pragma unroll 4
    for (int kk = 0; kk < 64; ++kk) {
      const int cbase = kk * 4 + 2 * half;  // ISA 16x4 f32 A/B fragment K index
      v2f bfrag = *(const v2f*)&Bsc[(wc * 16 + l15) * LSTRIDE + cbase];
      #pragma unroll
      for (int rt = 0; rt < 4; ++rt) {
        int row = (wr * 4 + rt) * 16 + l15;
        v2f afrag = *(const v2f*)&As[row * LSTRIDE + cbase];
        acc[rt] = __builtin_amdgcn_wmma_f32_16x16x4_f32(
            false, afrag, false, bfrag, (short)0, acc[rt], false, false);
      }
    }

    // Running argmin over chunks: score = ||e||^2 - 2*(z.e), packed sortable.
    #pragma unroll
    for (int rt = 0; rt < 4; ++rt) {
      #pragma unroll
      for (int i = 0; i < 8; ++i) {
        float s = ne - 2.0f * acc[rt][i];
        unsigned u = __float_as_uint(s);
        u = (u & 0x80000000u) ? ~u : (u | 0x80000000u);
        unsigned long long key = ((unsigned long long)u << 32) | vcol;
        best[rt][i] = key < best[rt][i] ? key : best[rt][i];
      }
    }

    if (wave == 0) __builtin_amdgcn_s_wait_tensorcnt(0);
    __syncthreads();   // next buffer ready; all waves done with current buffer
  }

  // Cross-lane then cross-block reduce (ties -> smallest v, like jnp.argmin).
  #pragma unroll
  for (int rt = 0; rt < 4; ++rt) {
    #pragma unroll
    for (int i = 0; i < 8; ++i) {
      unsigned long long k = best[rt][i];
      #pragma unroll
      for (int m = 8; m >= 1; m >>= 1) {
        unsigned long long o = __shfl_xor(k, m, 32);
        k = o < k ? o : k;
      }
      if (l15 == 0) {
        int row = mblk + (wr * 4 + rt) * 16 + 8 * half + i;  // D layout: M = i + 8*half
        atomicMin(&minslot[row], k);
      }
    }
  }
}

// ---------------- Kernel 2: indices out, z_q gather, loss partials ----------------
__global__ __launch_bounds__(256)
void vq_gather(const float* __restrict__ z,
               const float* __restrict__ cb,
               const unsigned long long* __restrict__ minslot,
               float* __restrict__ out,
               float* __restrict__ lossPart) {
  __shared__ int   sidx[16];
  __shared__ float sred[8];
  const int t  = threadIdx.x;
  const int m0 = blockIdx.x * 16;
  const int bb = m0 >> 10;
  const int n0 = m0 & (HW - 1);

  if (t < 16) {
    unsigned long long k = minslot[m0 + t];
    int v = (int)(unsigned)(k & 0xffffffffull);
    sidx[t] = v;
    out[OUT_IDX_OFF + m0 + t] = (float)v;
  }
  __syncthreads();

  const int r  = t & 15;
  const int cs = t >> 4;
  const int v  = sidx[r];
  const size_t zbase = (size_t)bb * EMBED * HW + n0 + r;

  float lsum = 0.f;
  #pragma unroll
  for (int it = 0; it < 16; ++it) {
    int c = it * 16 + cs;
    float zq = cb[(size_t)v * EMBED + c];
    size_t ga = zbase + (size_t)c * HW;
    float zv = z[ga];
    out[ga] = zq;
    float d = zq - zv;
    lsum += d * d;
  }
  #pragma unroll
  for (int m = 16; m >= 1; m >>= 1) lsum += __shfl_xor(lsum, m, 32);
  if ((t & 31) == 0) sred[t >> 5] = lsum;
  __syncthreads();
  if (t == 0) {
    float s = 0.f;
    #pragma unroll
    for (int w = 0; w < 8; ++w) s += sred[w];
    lossPart[blockIdx.x] = s;   // deterministic: no float atomics
  }
}

// ---------------- Kernel 3: perplexity + finalize scalars ----------------
__global__ __launch_bounds__(1024)
void vq_finalize(const float* __restrict__ lossPart,
                 float* __restrict__ out) {
  __shared__ float red[1024];
  const int t = threadIdx.x;
  const float* idxf = out + OUT_IDX_OFF;

  // Each distinct (n,v) with count c contributes -(c/8)log(c/8+eps),
  // spread as (1/8)log(...) per member entry.
  float h = 0.f;
  for (int e = t; e < MROWS; e += 1024) {
    int n = e & (HW - 1);
    int my = (int)idxf[e];
    int c = 0;
    #pragma unroll
    for (int b2 = 0; b2 < 8; ++b2)
      c += ((int)idxf[b2 * HW + n] == my) ? 1 : 0;
    float p = (float)c * 0.125f;
    h += 0.125f * logf(p + 1e-10f);
  }
  red[t] = h;
  __syncthreads();
  for (int s = 512; s >= 1; s >>= 1) {
    if (t < s) red[t] += red[t + s];
    __syncthreads();
  }
  float H = red[0];
  __syncthreads();

  red[t] = (t < 512) ? lossPart[t] : 0.f;
  __syncthreads();
  for (int s = 512; s >= 1; s >>= 1) {
    if (t < s) red[t] += red[t + s];
    __syncthreads();
  }
  if (t == 0) {
    out[OUT_LOSS_OFF] = red[0] / 2097152.0f;   // mean over B*C*H*W
    out[OUT_PERP_OFF] = expf(-H);
  }
}

extern "C" void kernel_launch(void* const* d_in, const int* in_sizes, int n_in,
                              void* d_out, int out_size, void* d_ws, size_t ws_size,
                              hipStream_t stream) {
  const float* z  = (const float*)d_in[0];   // (8, 256, 32, 32) f32
  const float* cb = (const float*)d_in[1];   // (8192, 256) f32
  float* out = (float*)d_out;
  char* ws = (char*)d_ws;
  float*              norm_e   = (float*)(ws + WS_NORM_OFF);
  unsigned long long* minslot  = (unsigned long long*)(ws + WS_MIN_OFF);
  float*              lossPart = (float*)(ws + WS_LOSSP_OFF);

  vq_init_normsq<<<dim3(VOCAB / 8), dim3(256), 0, stream>>>(cb, norm_e, minslot);
  vq_scores<<<dim3(MROWS / MT, VSLICES), dim3(256), 0, stream>>>(z, cb, norm_e, minslot);
  vq_gather<<<dim3(MROWS / 16), dim3(256), 0, stream>>>(z, cb, minslot, out, lossPart);
  vq_finalize<<<dim3(1), dim3(1024), 0, stream>>>(lossPart, out);
}